// VQCLayerAblation_1494648619143
// MI455X (gfx1250) — compile-verified
//
#include <hip/hip_runtime.h>

#define NQB 14
#define DIM 16384        // 2^14 amplitudes
#define NLAYERS 4
#define THREADS 256
#define NTILES 64        // 1024 cosets / 16 per WMMA tile

typedef __attribute__((ext_vector_type(2))) float v2f;
typedef __attribute__((ext_vector_type(8))) float v8f;

struct GroupDesc {
  int mask[4];   // XOR masks (columns of G_l, plus identity pads)
  int ins[4];    // sorted pivot bit positions for coset-rep enumeration
  int qubit[4];  // weight index, -1 => identity pad
};
struct Plan {
  GroupDesc grp[NLAYERS][4];
  int wmask[NQB];          // parity masks for Z-expectation (rows of G4^-1)
};

__global__ __launch_bounds__(THREADS) void vqc_wmma_kernel(
    const float* __restrict__ inputs,
    const float* __restrict__ weights,
    float* __restrict__ out,
    Plan plan)
{
  extern __shared__ float smem[];
  float* sre = smem;
  float* sim = smem + DIM;

  const int b      = blockIdx.x;
  const int tid    = threadIdx.x;
  const int lane   = tid & 31;
  const int waveId = tid >> 5;
  const int nWaves = THREADS >> 5;
  const int Mrow   = lane & 15;
  const int kb     = (lane >> 4) << 1;   // K sub-pair: 0 or 2

  // ---- Encoding: prod_q RY(x_q)|0> is a real product state; init directly ----
  float Cv[NQB], Sv[NQB];
  #pragma unroll
  for (int q = 0; q < NQB; ++q) {
    float x = inputs[b * NQB + q] * 0.5f;
    Cv[NQB - 1 - q] = __cosf(x);   // bitpos p = 13-q
    Sv[NQB - 1 - q] = __sinf(x);
  }
  for (int k = 0; k < DIM / THREADS; ++k) {
    int s = tid + (k << 8);        // stride-256: conflict-free banks
    float a = 1.0f;
    #pragma unroll
    for (int p = 0; p < NQB; ++p) a *= ((s >> p) & 1) ? Sv[p] : Cv[p];
    sre[s] = a;
    sim[s] = 0.0f;
  }

  // ---- Variational layers: CNOTs folded into masks; 4 fused 16x16 gates/layer ----
  for (int layer = 0; layer < NLAYERS; ++layer) {
    #pragma unroll
    for (int g = 0; g < 4; ++g) {
      const GroupDesc gd = plan.grp[layer][g];

      float cw[4], sw[4];
      #pragma unroll
      for (int k = 0; k < 4; ++k) {
        int q = gd.qubit[k];
        if (q >= 0) {
          float w = weights[layer * NQB + q] * 0.5f;
          cw[k] = __cosf(w); sw[k] = __sinf(w);
        } else { cw[k] = 1.0f; sw[k] = 0.0f; }
      }

      // A fragments: U = RX0 (x) RX1 (x) RX2 (x) RX3 in Kron form:
      // U[r][c] = (-i)^popc(r^c) * prod_k ((r^c)_k ? sw_k : cw_k)
      v2f Ar[4], Ai[4], Ain[4];
      #pragma unroll
      for (int kc = 0; kc < 4; ++kc) {
        #pragma unroll
        for (int j = 0; j < 2; ++j) {
          int c = 4 * kc + kb + j;
          int x = Mrow ^ c;
          float mag = 1.0f;
          #pragma unroll
          for (int k = 0; k < 4; ++k) mag *= ((x >> k) & 1) ? sw[k] : cw[k];
          int n = __popc(x) & 3;
          float ur = (n == 0) ? mag : ((n == 2) ? -mag : 0.0f);
          float ui = (n == 1) ? -mag : ((n == 3) ? mag : 0.0f);
          Ar[kc][j] = ur; Ai[kc][j] = ui; Ain[kc][j] = -ui;
        }
      }

      __syncthreads();   // previous sweep fully written

      for (int t = waveId; t < NTILES; t += nWaves) {
        int blk = (t << 4) + Mrow;           // coset id (column N of tile)
        int s0 = blk;                         // coset representative: insert 0s at pivots
        #pragma unroll
        for (int k = 0; k < 4; ++k) {
          int p = gd.ins[k];
          s0 = ((s0 >> p) << (p + 1)) | (s0 & ((1 << p) - 1));
        }
        // Gather B fragments (K = coset element, N = coset id)
        v2f Br[4], Bi[4];
        #pragma unroll
        for (int kc = 0; kc < 4; ++kc) {
          #pragma unroll
          for (int j = 0; j < 2; ++j) {
            int e  = 4 * kc + kb + j;
            int xm = ((e & 1) ? gd.mask[0] : 0) ^ ((e & 2) ? gd.mask[1] : 0)
                   ^ ((e & 4) ? gd.mask[2] : 0) ^ ((e & 8) ? gd.mask[3] : 0);
            int idx = s0 ^ xm;
            Br[kc][j] = sre[idx];
            Bi[kc][j] = sim[idx];
          }
        }
        // Complex D = U * X via 16 f32 WMMAs (K=4 chunks x 4 real products)
        v8f dr = {0.f,0.f,0.f,0.f,0.f,0.f,0.f,0.f};
        v8f di = {0.f,0.f,0.f,0.f,0.f,0.f,0.f,0.f};
        #pragma unroll
        for (int kc = 0; kc < 4; ++kc) {
          dr = __builtin_amdgcn_wmma_f32_16x16x4_f32(false, Ar[kc],  false, Br[kc], (short)0, dr, false, false);
          dr = __builtin_amdgcn_wmma_f32_16x16x4_f32(false, Ain[kc], false, Bi[kc], (short)0, dr, false, false);
          di = __builtin_amdgcn_wmma_f32_16x16x4_f32(false, Ar[kc],  false, Bi[kc], (short)0, di, false, false);
          di = __builtin_amdgcn_wmma_f32_16x16x4_f32(false, Ai[kc],  false, Br[kc], (short)0, di, false, false);
        }
        // Scatter D back (lane holds rows M = 8*(lane>=16)+v of column N)
        #pragma unroll
        for (int v = 0; v < 8; ++v) {
          int Mo = ((lane >> 4) << 3) + v;
          int xm = ((Mo & 1) ? gd.mask[0] : 0) ^ ((Mo & 2) ? gd.mask[1] : 0)
                 ^ ((Mo & 4) ? gd.mask[2] : 0) ^ ((Mo & 8) ? gd.mask[3] : 0);
          int idx = s0 ^ xm;
          sre[idx] = dr[v];
          sim[idx] = di[v];
        }
      }
    }
  }
  __syncthreads();

  // ---- Z expectations with folded-CNOT parity masks ----
  float acc[NQB];
  #pragma unroll
  for (int i = 0; i < NQB; ++i) acc[i] = 0.0f;
  for (int k = 0; k < DIM / THREADS; ++k) {
    int s = tid + (k << 8);
    float re = sre[s], im = sim[s];
    float p = re * re + im * im;
    #pragma unroll
    for (int i = 0; i < NQB; ++i)
      acc[i] += (__popc(s & plan.wmask[i]) & 1) ? -p : p;
  }
  __syncthreads();
  #pragma unroll
  for (int i = 0; i < NQB; ++i) smem[i * THREADS + tid] = acc[i];
  __syncthreads();
  if (tid < NQB) {
    float sum = 0.0f;
    for (int t = 0; t < THREADS; ++t) sum += smem[tid * THREADS + t];
    out[b * NQB + tid] = sum;   // deterministic serial reduction
  }
}

// ===================== host-side GF(2) plan construction =====================
struct LinMap { int col[NQB]; };

static LinMap lm_identity() { LinMap m; for (int p = 0; p < NQB; ++p) m.col[p] = 1 << p; return m; }
static int lm_apply(const LinMap& m, int x) {
  int r = 0; for (int p = 0; p < NQB; ++p) if ((x >> p) & 1) r ^= m.col[p]; return r;
}
static LinMap lm_compose(const LinMap& a, const LinMap& b) {   // (a o b)(y) = a(b(y))
  LinMap r; for (int p = 0; p < NQB; ++p) r.col[p] = lm_apply(a, b.col[p]); return r;
}
static LinMap lm_cnot(int pc, int pt) { LinMap m = lm_identity(); m.col[pc] ^= (1 << pt); return m; }

static bool independentOf(const int* arr, int n, int cand) {
  int em[8], piv[8]; int ne = 0;
  for (int i = 0; i < n; ++i) {
    int r = arr[i];
    for (int j = 0; j < ne; ++j) if ((r >> piv[j]) & 1) r ^= em[j];
    if (!r) continue;
    int p = 0; for (int bb = NQB - 1; bb >= 0; --bb) if ((r >> bb) & 1) { p = bb; break; }
    em[ne] = r; piv[ne] = p; ++ne;
  }
  int r = cand;
  for (int j = 0; j < ne; ++j) if ((r >> piv[j]) & 1) r ^= em[j];
  return r != 0;
}

static void buildPlan(Plan& plan) {
  // Layer CNOT chain i=0..12 applied in order => psi'[y] = psi[H(y)], H = f01 o f12 o ... o f(12,13)
  LinMap H = lm_identity();
  for (int i = 12; i >= 0; --i) H = lm_compose(lm_cnot(13 - i, 12 - i), H);
  LinMap Hinv = lm_identity();
  for (int i = 0; i <= 12; ++i) Hinv = lm_compose(lm_cnot(13 - i, 12 - i), Hinv);

  LinMap G = lm_identity();
  for (int l = 0; l < NLAYERS; ++l) {
    G = lm_compose(G, H);                      // G_l = H^l
    for (int g = 0; g < 4; ++g) {
      GroupDesc& gd = plan.grp[l][g];
      int nqg = (g < 3) ? 4 : 2;
      int ms[4], qs[4];
      for (int k = 0; k < nqg; ++k) { int q = 4 * g + k; qs[k] = q; ms[k] = G.col[13 - q]; }
      int nb = nqg;                            // pad with independent unit masks + identity gates
      for (int cand = 0; cand < NQB && nb < 4; ++cand)
        if (independentOf(ms, nb, 1 << cand)) { ms[nb] = 1 << cand; qs[nb] = -1; ++nb; }
      // echelon pivots -> coset-rep insertion positions
      int piv[4], em[4];
      for (int k = 0; k < 4; ++k) {
        int r = ms[k];
        for (int j = 0; j < k; ++j) if ((r >> piv[j]) & 1) r ^= em[j];
        int p = 0; for (int bb = NQB - 1; bb >= 0; --bb) if ((r >> bb) & 1) { p = bb; break; }
        piv[k] = p; em[k] = r;
      }
      int sp[4] = { piv[0], piv[1], piv[2], piv[3] };
      for (int a = 0; a < 4; ++a)
        for (int c = a + 1; c < 4; ++c)
          if (sp[c] < sp[a]) { int t = sp[a]; sp[a] = sp[c]; sp[c] = t; }
      for (int k = 0; k < 4; ++k) { gd.mask[k] = ms[k]; gd.ins[k] = sp[k]; gd.qubit[k] = qs[k]; }
    }
  }
  LinMap Ginv = lm_identity();
  for (int l = 0; l < NLAYERS; ++l) Ginv = lm_compose(Ginv, Hinv);   // G4^-1
  for (int i = 0; i < NQB; ++i) {
    int w = 0;
    for (int p = 0; p < NQB; ++p) if ((Ginv.col[p] >> (13 - i)) & 1) w |= (1 << p);
    plan.wmask[i] = w;
  }
}

extern "C" void kernel_launch(void* const* d_in, const int* in_sizes, int n_in,
                              void* d_out, int out_size, void* d_ws, size_t ws_size,
                              hipStream_t stream) {
  (void)in_sizes; (void)n_in; (void)d_ws; (void)ws_size; (void)out_size;
  const float* inputs  = (const float*)d_in[0];   // (1024, 14) f32
  const float* weights = (const float*)d_in[1];   // (4, 14)   f32
  float* out = (float*)d_out;                     // (1024, 14) f32

  Plan plan;
  buildPlan(plan);

  vqc_wmma_kernel<<<1024, THREADS, 2 * DIM * sizeof(float), stream>>>(inputs, weights, out, plan);
}